// ZBLRepulsion_59622736003517
// MI455X (gfx1250) — compile-verified
//
#include <hip/hip_runtime.h>
#include <stdint.h>

#define N_ATOMS 100000
#define N_EDGES 6400000
#define TPB 256
#define EPT 4                       // edges per thread per tile (one int4 per stream)
#define TILE (TPB * EPT)            // 1024 edges per tile
#define NTILES (N_EDGES / TILE)     // 6250
#define NBLK 1250
#define TILES_PER_BLK (NTILES / NBLK) // 5
#define KE_CONST 14.3996f
#define LOG2E 1.44269504088896f

// workspace layout (float units)
#define WS_TABLE   0                      // 64-entry Z^ae table
#define WS_CONST   64                     // 16 scalar constants
#define WS_PACKED  96                     // float4 per atom (byte offset 384, 16B aligned)
#define WS_PARTIAL (WS_PACKED + 4 * N_ATOMS)

__device__ __forceinline__ float softplus_f(float x) { return log1pf(expf(x)); }

// gfx1250 async global->LDS copy (ASYNCcnt-tracked), saddr form:
//   global_load_async_to_lds_b128 vdst(lds byte addr), vaddr(u32 off), s[base:base+1]
__device__ __forceinline__ void async_copy_b128(uint32_t lds_addr, uint32_t voff,
                                                const void* base) {
    asm volatile("global_load_async_to_lds_b128 %0, %1, %2"
                 :: "v"(lds_addr), "v"(voff), "s"(base)
                 : "memory");
}

__global__ void zbl_setup_kernel(const float* __restrict__ a_exp,
                                 const float* __restrict__ a_num,
                                 const float* __restrict__ coef,
                                 const float* __restrict__ expo,
                                 const float* __restrict__ rep,
                                 float* __restrict__ ws) {
    int t = threadIdx.x;
    float ae = softplus_f(a_exp[0]);
    if (t < 64) {
        float z = (float)t;
        // z^ae = exp2(ae*log2(z)); z=0 -> exp2(-inf) = 0 (unused, Z >= 1)
        // __builtin_amdgcn_logf == v_log_f32 == log2
        ws[WS_TABLE + t] = __builtin_amdgcn_exp2f(ae * __builtin_amdgcn_logf(z));
    }
    if (t == 0) {
        float* c = ws + WS_CONST;
        for (int k = 0; k < 4; ++k) c[k] = softplus_f(coef[k]);
        // fold exp(-e*d) into exp2(ek2*d) with ek2 = -e*log2(e)
        for (int k = 0; k < 4; ++k) c[4 + k] = -softplus_f(expo[k]) * LOG2E;
        c[8] = 1.0f / softplus_f(a_num[0]);
        c[9] = 0.5f * softplus_f(rep[0]) * KE_CONST;   // final scale
    }
}

__global__ void zbl_pack_kernel(const float* __restrict__ R,
                                const int* __restrict__ Z,
                                float* __restrict__ ws) {
    int a = blockIdx.x * blockDim.x + threadIdx.x;
    if (a < N_ATOMS) {
        float4 v;
        v.x = R[3 * a + 0];
        v.y = R[3 * a + 1];
        v.z = R[3 * a + 2];
        v.w = (float)Z[a];
        ((float4*)(ws + WS_PACKED))[a] = v;
    }
}

__device__ __forceinline__ float edge_energy(int ii, int jj,
                                             const float4* __restrict__ packed,
                                             const float* __restrict__ sPow,
                                             float c0, float c1, float c2, float c3,
                                             float e0, float e1, float e2, float e3,
                                             float an_inv) {
    if (ii == jj) return 0.0f;                     // reference mask
    float4 ai = packed[ii];
    float4 aj = packed[jj];
    float dx = aj.x - ai.x, dy = aj.y - ai.y, dz = aj.z - ai.z;
    float r2 = fmaf(dx, dx, fmaf(dy, dy, dz * dz));
    float dr = __builtin_amdgcn_sqrtf(r2);
    dr = fminf(fmaxf(dr, 0.02f), 6.0f);
    float inv_dr = __builtin_amdgcn_rcpf(dr);
    float pi_ = sPow[(int)ai.w];
    float pj_ = sPow[(int)aj.w];
    float dist = dr * (pi_ + pj_) * an_inv;
    float f = c0 * __builtin_amdgcn_exp2f(e0 * dist)
            + c1 * __builtin_amdgcn_exp2f(e1 * dist)
            + c2 * __builtin_amdgcn_exp2f(e2 * dist)
            + c3 * __builtin_amdgcn_exp2f(e3 * dist);
    // cos(pi*dr/6) == v_cos(dr/12): v_cos input is in revolutions
    float cut = 0.5f * (__builtin_amdgcn_cosf(dr * (1.0f / 12.0f)) + 1.0f);
    return ai.w * aj.w * inv_dr * f * cut;
}

__global__ void __launch_bounds__(TPB)
zbl_edge_kernel(const int* __restrict__ idx, float* __restrict__ ws) {
    __shared__ int4  sI[2][TPB];
    __shared__ int4  sJ[2][TPB];
    __shared__ float sPow[64];
    __shared__ float sRed[TPB / 32];

    const int tid = threadIdx.x;
    if (tid < 64) sPow[tid] = ws[WS_TABLE + tid];
    __syncthreads();

    const float* cst = ws + WS_CONST;
    const float c0 = cst[0], c1 = cst[1], c2 = cst[2], c3 = cst[3];
    const float e0 = cst[4], e1 = cst[5], e2 = cst[6], e3 = cst[7];
    const float an_inv = cst[8];
    const float4* __restrict__ packed = (const float4*)(ws + WS_PACKED);

    const int* idx_i = idx;
    const int* idx_j = idx + N_EDGES;

    // low 32 bits of a generic LDS pointer == wave LDS byte offset
    const uint32_t ldsI[2] = {(uint32_t)(uintptr_t)&sI[0][tid],
                              (uint32_t)(uintptr_t)&sI[1][tid]};
    const uint32_t ldsJ[2] = {(uint32_t)(uintptr_t)&sJ[0][tid],
                              (uint32_t)(uintptr_t)&sJ[1][tid]};

    const int tileBase = blockIdx.x * TILES_PER_BLK;

    // prime the pipeline: stage tile 0 into buffer 0
    {
        uint32_t off = (uint32_t)((tileBase * TILE + tid * EPT) * 4);
        async_copy_b128(ldsI[0], off, idx_i);
        async_copy_b128(ldsJ[0], off, idx_j);
    }

    float acc = 0.0f;
    for (int t = 0; t < TILES_PER_BLK; ++t) {
        const int buf = t & 1;
        if (t + 1 < TILES_PER_BLK) {
            uint32_t off = (uint32_t)(((tileBase + t + 1) * TILE + tid * EPT) * 4);
            async_copy_b128(ldsI[buf ^ 1], off, idx_i);
            async_copy_b128(ldsJ[buf ^ 1], off, idx_j);
            // 2 async ops for next tile in flight; wait until current tile's 2 are done
            asm volatile("s_wait_asynccnt 2" ::: "memory");
        } else {
            asm volatile("s_wait_asynccnt 0" ::: "memory");
        }
        int4 vi = sI[buf][tid];   // each thread consumes only its own staged slot
        int4 vj = sJ[buf][tid];
        acc += edge_energy(vi.x, vj.x, packed, sPow, c0, c1, c2, c3, e0, e1, e2, e3, an_inv);
        acc += edge_energy(vi.y, vj.y, packed, sPow, c0, c1, c2, c3, e0, e1, e2, e3, an_inv);
        acc += edge_energy(vi.z, vj.z, packed, sPow, c0, c1, c2, c3, e0, e1, e2, e3, an_inv);
        acc += edge_energy(vi.w, vj.w, packed, sPow, c0, c1, c2, c3, e0, e1, e2, e3, an_inv);
    }

    // wave32 butterfly reduce, then cross-wave via LDS
    for (int o = 16; o > 0; o >>= 1) acc += __shfl_xor(acc, o, 32);
    const int wave = tid >> 5, lane = tid & 31;
    if (lane == 0) sRed[wave] = acc;
    __syncthreads();
    if (tid == 0) {
        float s = 0.0f;
        for (int w = 0; w < TPB / 32; ++w) s += sRed[w];
        ws[WS_PARTIAL + blockIdx.x] = s;
    }
}

__global__ void __launch_bounds__(TPB)
zbl_final_kernel(const float* __restrict__ ws, float* __restrict__ out) {
    __shared__ float sRed[TPB / 32];
    const int tid = threadIdx.x;
    float acc = 0.0f;
    for (int p = tid; p < NBLK; p += TPB) acc += ws[WS_PARTIAL + p];
    for (int o = 16; o > 0; o >>= 1) acc += __shfl_xor(acc, o, 32);
    const int wave = tid >> 5, lane = tid & 31;
    if (lane == 0) sRed[wave] = acc;
    __syncthreads();
    if (tid == 0) {
        float s = 0.0f;
        for (int w = 0; w < TPB / 32; ++w) s += sRed[w];
        out[0] = ws[WS_CONST + 9] * s;   // 0.5 * rs * KE * sum
    }
}

extern "C" void kernel_launch(void* const* d_in, const int* in_sizes, int n_in,
                              void* d_out, int out_size, void* d_ws, size_t ws_size,
                              hipStream_t stream) {
    const float* R     = (const float*)d_in[0];
    const int*   Z     = (const int*)d_in[1];
    const int*   idx   = (const int*)d_in[2];
    // d_in[3] = box, d_in[4] = offsets: unused by the reference math
    const float* a_exp = (const float*)d_in[5];
    const float* a_num = (const float*)d_in[6];
    const float* coef  = (const float*)d_in[7];
    const float* expo  = (const float*)d_in[8];
    const float* rep   = (const float*)d_in[9];
    float* ws  = (float*)d_ws;
    float* out = (float*)d_out;

    zbl_setup_kernel<<<1, 64, 0, stream>>>(a_exp, a_num, coef, expo, rep, ws);
    zbl_pack_kernel<<<(N_ATOMS + TPB - 1) / TPB, TPB, 0, stream>>>(R, Z, ws);
    zbl_edge_kernel<<<NBLK, TPB, 0, stream>>>(idx, ws);
    zbl_final_kernel<<<1, TPB, 0, stream>>>(ws, out);
}